// ChannelAttention_64493228916840
// MI455X (gfx1250) — compile-verified
//
#include <hip/hip_runtime.h>
#include <math.h>

typedef float v2f __attribute__((ext_vector_type(2)));
typedef float v8f __attribute__((ext_vector_type(8)));

#define CH 64
#define TILE_N 128
#define VSTRIDE 132     // LDS row stride for 64x128 tile: bank = (4*row + col) % 64
#define MSTRIDE 68      // LDS row stride for 64x64 matrix
#define EPS 1e-6f

__global__ void zero_ws_kernel(float* __restrict__ ws, int n) {
    int i = blockIdx.x * blockDim.x + threadIdx.x;
    if (i < n) ws[i] = 0.0f;
}

// ---------------------------------------------------------------------------
// Pass 1: per batch accumulate matrix[64][64] = V * Qn^T  and  t[64]
// grid: (wgPerBatch, B), block: 256 threads = 8 waves (wave32)
// ---------------------------------------------------------------------------
__global__ __launch_bounds__(256)
void pass1_kernel(const float* __restrict__ x, float* __restrict__ wsM,
                  float* __restrict__ wsT, int N, int colsPerWG)
{
    const int b    = blockIdx.y;
    const int tid  = threadIdx.x;
    const int wave = tid >> 5;
    const int lane = tid & 31;

    __shared__ float tileV[CH * VSTRIDE];
    __shared__ float invn[TILE_N];
    __shared__ float wcol[TILE_N];   // s[n] * inv[n]^2

    const float* xb = x + (size_t)b * CH * N;

    // wave owns two (c-tile, m-tile) outputs: same c-tile, two adjacent m-tiles
    const int c0  = (wave >> 1) * 16;
    const int m0a = ((wave & 1) * 2 + 0) * 16;
    const int m0b = ((wave & 1) * 2 + 1) * 16;

    const int rA   = lane & 15;            // M (or N) index within tile
    const int kOff = (lane < 16) ? 0 : 2;  // K sub-offset for this half-wave

    v8f accA = {};
    v8f accB = {};
    float tacc = 0.0f;

    const int colStart = blockIdx.x * colsPerWG;
    for (int base = colStart; base < colStart + colsPerWG; base += TILE_N) {
        // ---- cooperative coalesced load: wave w loads rows 8w..8w+7 ----
        {
            const int r0 = wave * 8;
            #pragma unroll
            for (int rr = 0; rr < 8; ++rr) {
                const int r = r0 + rr;
                float4 val = *(const float4*)(xb + (size_t)r * N + base + lane * 4);
                *(float4*)&tileV[r * VSTRIDE + lane * 4] = val;
            }
        }
        __syncthreads();

        // ---- per-column stats (threads 0..127, one column each) ----
        if (tid < TILE_N) {
            float ss = 0.0f, s = 0.0f;
            #pragma unroll 8
            for (int c = 0; c < CH; ++c) {
                float v = tileV[c * VSTRIDE + tid];
                ss += v * v;
                s  += v;
            }
            float inv = 1.0f / (sqrtf(ss) + EPS);
            invn[tid] = inv;
            wcol[tid] = s * inv * inv;
        }
        __syncthreads();

        // ---- t[c] partial: thread covers channel (tid&63), column quarter ----
        {
            const int c = tid & 63;
            const int q = tid >> 6;
            const float* row = &tileV[c * VSTRIDE + q * 32];
            const float* w   = &wcol[q * 32];
            float a = 0.0f;
            #pragma unroll 8
            for (int j = 0; j < 32; ++j) a += row[j] * w[j];
            tacc += a;
        }

        // ---- WMMA: D[c,m] += A(V rows) x B(Qn rows), K steps of 4 ----
        #pragma unroll 4
        for (int k0 = 0; k0 < TILE_N; k0 += 4) {
            const float i0 = invn[k0 + kOff];
            const float i1 = invn[k0 + kOff + 1];
            v2f a, bA, bB;
            a.x  = tileV[(c0  + rA) * VSTRIDE + k0 + kOff];
            a.y  = tileV[(c0  + rA) * VSTRIDE + k0 + kOff + 1];
            bA.x = tileV[(m0a + rA) * VSTRIDE + k0 + kOff]     * i0;
            bA.y = tileV[(m0a + rA) * VSTRIDE + k0 + kOff + 1] * i1;
            bB.x = tileV[(m0b + rA) * VSTRIDE + k0 + kOff]     * i0;
            bB.y = tileV[(m0b + rA) * VSTRIDE + k0 + kOff + 1] * i1;
            accA = __builtin_amdgcn_wmma_f32_16x16x4_f32(false, a, false, bA,
                                                         (short)0, accA, false, false);
            accB = __builtin_amdgcn_wmma_f32_16x16x4_f32(false, a, false, bB,
                                                         (short)0, accB, false, false);
        }
        __syncthreads();
    }

    // ---- flush accumulators: C/D layout: VGPR r -> M = r (lanes<16) / r+8 ----
    {
        float* Mb = wsM + (size_t)b * CH * CH;
        const int nloc  = lane & 15;
        const int mbase = (lane < 16) ? 0 : 8;
        #pragma unroll
        for (int r = 0; r < 8; ++r) {
            atomicAdd(&Mb[(c0 + mbase + r) * CH + m0a + nloc], accA[r]);
            atomicAdd(&Mb[(c0 + mbase + r) * CH + m0b + nloc], accB[r]);
        }
        atomicAdd(&wsT[b * CH + (tid & 63)], tacc);
    }
}

// ---------------------------------------------------------------------------
// Pass 2: out = x + gamma * tailor[c] * (s[n] + matrix @ qn)
// grid: (wgPerBatch, B), block: 256 threads = 8 waves
// ---------------------------------------------------------------------------
__global__ __launch_bounds__(256)
void pass2_kernel(const float* __restrict__ x, const float* __restrict__ gamma_p,
                  const float* __restrict__ wsM, const float* __restrict__ wsT,
                  float* __restrict__ out, int N, int colsPerWG)
{
    const int b    = blockIdx.y;
    const int tid  = threadIdx.x;
    const int wave = tid >> 5;
    const int lane = tid & 31;

    __shared__ float mat[CH * MSTRIDE];
    __shared__ float tileV[CH * VSTRIDE];
    __shared__ float invn[TILE_N];
    __shared__ float scol[TILE_N];
    __shared__ float tail[CH];

    const float gamma = gamma_p[0];
    const float* xb = x + (size_t)b * CH * N;
    float* ob = out + (size_t)b * CH * N;

    // stage matrix + tailor in LDS
    for (int i = tid; i < CH * CH; i += 256)
        mat[(i >> 6) * MSTRIDE + (i & 63)] = wsM[(size_t)b * CH * CH + i];
    if (tid < CH)
        tail[tid] = 1.0f / ((float)N + wsT[b * CH + tid] + EPS);
    __syncthreads();

    const int rA   = lane & 15;
    const int kOff = (lane < 16) ? 0 : 2;

    const int colStart = blockIdx.x * colsPerWG;
    for (int base = colStart; base < colStart + colsPerWG; base += TILE_N) {
        // ---- load 64x128 tile (coalesced) ----
        {
            const int r0 = wave * 8;
            #pragma unroll
            for (int rr = 0; rr < 8; ++rr) {
                const int r = r0 + rr;
                float4 val = *(const float4*)(xb + (size_t)r * N + base + lane * 4);
                *(float4*)&tileV[r * VSTRIDE + lane * 4] = val;
            }
        }
        __syncthreads();

        if (tid < TILE_N) {
            float ss = 0.0f, s = 0.0f;
            #pragma unroll 8
            for (int c = 0; c < CH; ++c) {
                float v = tileV[c * VSTRIDE + tid];
                ss += v * v;
                s  += v;
            }
            invn[tid] = 1.0f / (sqrtf(ss) + EPS);
            scol[tid] = s;
        }
        __syncthreads();

        // ---- 32 output tiles (4 c-tiles x 8 n-tiles); 4 per wave ----
        #pragma unroll
        for (int i = 0; i < 4; ++i) {
            const int t  = wave * 4 + i;
            const int c0 = (t >> 3) * 16;
            const int n0 = (t & 7) * 16;

            const float iv = invn[n0 + rA];   // per-lane column scale (const over K)
            v8f acc = {};
            #pragma unroll
            for (int k0 = 0; k0 < CH; k0 += 4) {
                v2f a, bf;
                a.x  = mat[(c0 + rA) * MSTRIDE + k0 + kOff];
                a.y  = mat[(c0 + rA) * MSTRIDE + k0 + kOff + 1];
                bf.x = tileV[(k0 + kOff)     * VSTRIDE + n0 + rA] * iv;
                bf.y = tileV[(k0 + kOff + 1) * VSTRIDE + n0 + rA] * iv;
                acc = __builtin_amdgcn_wmma_f32_16x16x4_f32(false, a, false, bf,
                                                            (short)0, acc, false, false);
            }

            // ---- fused epilogue + store ----
            const int nloc  = lane & 15;
            const int mbase = (lane < 16) ? 0 : 8;
            const float sv  = scol[n0 + nloc];
            const int   n   = base + n0 + nloc;
            #pragma unroll
            for (int r = 0; r < 8; ++r) {
                const int c = c0 + mbase + r;
                const float xv = tileV[c * VSTRIDE + n0 + nloc];
                ob[(size_t)c * N + n] = xv + gamma * tail[c] * (sv + acc[r]);
            }
        }
        __syncthreads();
    }
}

// ---------------------------------------------------------------------------
extern "C" void kernel_launch(void* const* d_in, const int* in_sizes, int n_in,
                              void* d_out, int out_size, void* d_ws, size_t ws_size,
                              hipStream_t stream) {
    const float* x     = (const float*)d_in[0];
    const float* gamma = (const float*)d_in[1];
    float* out = (float*)d_out;
    float* ws  = (float*)d_ws;

    const int B = 16, C = 64, N = 256 * 256;

    float* wsM = ws;                        // B * C * C
    float* wsT = ws + (size_t)B * C * C;    // B * C
    const int wsTotal = B * C * C + B * C;  // 66560 floats (~260 KB)

    zero_ws_kernel<<<(wsTotal + 255) / 256, 256, 0, stream>>>(ws, wsTotal);

    const int wg1 = 32;                      // 32 WGs/batch -> 512 WGs
    pass1_kernel<<<dim3(wg1, B), 256, 0, stream>>>(x, wsM, wsT, N, N / wg1);

    const int wg2 = 128;                     // 128 WGs/batch -> 2048 WGs
    pass2_kernel<<<dim3(wg2, B), 256, 0, stream>>>(x, gamma, wsM, wsT, out, N, N / wg2);
}